// GCNPolicy_10574209483591
// MI455X (gfx1250) — compile-verified
//
#include <hip/hip_runtime.h>

#define OBS 128
#define HID 128
#define ACT 8
#define NGRAPHS 1024

typedef float v2f __attribute__((ext_vector_type(2)));
typedef float v8f __attribute__((ext_vector_type(8)));
typedef unsigned int u32x4 __attribute__((ext_vector_type(4)));
typedef int i32x4 __attribute__((ext_vector_type(4)));
typedef int i32x8 __attribute__((ext_vector_type(8)));

__device__ __forceinline__ v8f wmma_f32_4(v2f a, v2f b, v8f c) {
  // D = A(16x4 f32) x B(4x16 f32) + C(16x16 f32)
  return __builtin_amdgcn_wmma_f32_16x16x4_f32(false, a, false, b, (short)0, c,
                                               false, false);
}

// ---------------------------------------------------------------------------
// TDM: DMA a 2D f32 tile (tile_d1 rows x tile_d0 cols) from global into LDS.
// Columns past tensor_d0 are hardware zero-filled (OOB reads return zero),
// which we exploit to zero-pad the head weight matrix. Issue from one wave,
// then s_wait_tensorcnt 0 before the workgroup barrier.
// ---------------------------------------------------------------------------
__device__ __forceinline__ void tdm_load_2d_f32(unsigned lds_off,
                                                const void* gptr,
                                                unsigned tensor_d0,
                                                unsigned tensor_d1,
                                                unsigned tile_d0,
                                                unsigned tile_d1,
                                                unsigned stride0) {
  unsigned long long ga = (unsigned long long)(uintptr_t)gptr;
  u32x4 g0;
  g0.x = 1u;                                             // count=1 (valid D#)
  g0.y = lds_off;                                        // lds_addr [63:32]
  g0.z = (unsigned)ga;                                   // global_addr[31:0]
  g0.w = (unsigned)((ga >> 32) & 0x1FFFFFFu) | (2u << 30);  // addr[56:32]|type=2
  i32x8 g1;
  g1[0] = (int)(2u << 16);                   // data_size=2 (4 bytes)
  g1[1] = (int)(tensor_d0 << 16);            // tensor_dim0[15:0] @ bits 63:48
  g1[2] = (int)((tensor_d0 >> 16) | (tensor_d1 << 16));
  g1[3] = (int)((tensor_d1 >> 16) | (tile_d0 << 16));
  g1[4] = (int)(tile_d1 & 0xFFFFu);          // tile_dim1; tile_dim2=0
  g1[5] = (int)stride0;                      // tensor_dim0_stride[31:0]
  g1[6] = 0;                                 // stride0[47:32]=0; stride1 lo=0
  g1[7] = 0;                                 // stride1 hi=0
  i32x4 z4 = {0, 0, 0, 0};                   // 2D: groups 2/3 unused
  i32x8 z8 = {0, 0, 0, 0, 0, 0, 0, 0};
  __builtin_amdgcn_tensor_load_to_lds(g0, g1, z4, z4, z8, 0);
}

// ---------------------------------------------------------------------------
// Utility kernels
// ---------------------------------------------------------------------------
__global__ void zero_kernel(float* __restrict__ p, int n) {
  int i = blockIdx.x * blockDim.x + threadIdx.x;
  if (i < n) p[i] = 0.0f;
}

__global__ void fill_one_kernel(float* __restrict__ p, int n) {
  int i = blockIdx.x * blockDim.x + threadIdx.x;
  if (i < n) p[i] = 1.0f;  // self-loop contributes 1 to in-degree
}

__global__ void count_deg_kernel(const int* __restrict__ dst,
                                 float* __restrict__ deg, int E) {
  int e = blockIdx.x * blockDim.x + threadIdx.x;
  if (e < E) atomicAdd(&deg[dst[e]], 1.0f);
}

__global__ void deg_rsqrt_kernel(float* __restrict__ deg, int n) {
  int i = blockIdx.x * blockDim.x + threadIdx.x;
  if (i < n) deg[i] = rsqrtf(deg[i]);  // deg >= 1 always (self-loop)
}

__global__ void count_nodes_kernel(const int* __restrict__ batch,
                                   float* __restrict__ cnt, int n) {
  int i = blockIdx.x * blockDim.x + threadIdx.x;
  if (i < n) atomicAdd(&cnt[batch[i]], 1.0f);
}

// ---------------------------------------------------------------------------
// Dense GEMM: Y[N,128] = X[N,128] @ W[128,128] via V_WMMA_F32_16X16X4_F32
// 256 threads = 8 waves; block covers 128 rows; W staged in 64KB LDS by TDM.
// k-outer / ct-inner: one A fragment feeds 8 WMMAs (8 independent acc chains).
// A rows are clamped (not predicated); stores take an unguarded fast path
// when the whole 16-row tile is in range (wave-uniform test).
// ---------------------------------------------------------------------------
__global__ void __launch_bounds__(256)
gemm128_wmma(const float* __restrict__ X, const float* __restrict__ W,
             float* __restrict__ Y, int N) {
  __shared__ float sW[128 * 128];  // 64 KB
  const int t = threadIdx.x;
  if (t == 0) {
    tdm_load_2d_f32((unsigned)(uintptr_t)sW, W, 128, 128, 128, 128, 128);
    __builtin_amdgcn_s_wait_tensorcnt(0);
  }
  __syncthreads();

  const int wave = t >> 5;
  const int lane = t & 31;
  const int r = lane & 15;
  const int khalf = (lane >> 4) << 1;  // lanes 16..31 hold K=+2,+3
  const int rowbase = blockIdx.x * 128 + wave * 16;
  int row = rowbase + r;
  row = row < N ? row : N - 1;  // clamp; stores below handle rr >= N
  const float* xrow = X + (size_t)row * HID;

  v8f acc[8];
  {
    v8f z = {};
#pragma unroll
    for (int ct = 0; ct < 8; ++ct) acc[ct] = z;
  }

#pragma unroll 4
  for (int k = 0; k < 128; k += 4) {
    const int kk = k + khalf;
    float2 av = *(const float2*)(xrow + kk);
    v2f a;
    a.x = av.x;
    a.y = av.y;
#pragma unroll
    for (int ct = 0; ct < 8; ++ct) {
      v2f b;
      b.x = sW[kk * 128 + ct * 16 + r];
      b.y = sW[(kk + 1) * 128 + ct * 16 + r];
      acc[ct] = wmma_f32_4(a, b, acc[ct]);
    }
  }

  // C layout: VGPR j -> row = base + (lane>=16 ? 8:0) + j, col = lane&15
  const int mbase = rowbase + ((lane >> 4) << 3);
  if (rowbase + 16 <= N) {  // full tile in range (wave-uniform fast path)
    float* yb = Y + (size_t)mbase * HID + r;
#pragma unroll
    for (int ct = 0; ct < 8; ++ct) {
#pragma unroll
      for (int j = 0; j < 8; ++j) yb[(size_t)j * HID + ct * 16] = acc[ct][j];
    }
  } else if (rowbase < N) {  // partial tail tile
#pragma unroll
    for (int ct = 0; ct < 8; ++ct) {
      const int col = ct * 16 + r;
#pragma unroll
      for (int j = 0; j < 8; ++j) {
        const int rr = mbase + j;
        if (rr < N) Y[(size_t)rr * HID + col] = acc[ct][j];
      }
    }
  }
}

// ---------------------------------------------------------------------------
// Edge scatter: agg[dst] += h[src] * dinv[src]*dinv[dst]
// One wave32 per edge; each lane moves a float4 (32*16B = 512B = 128 floats).
// h and agg are L2-resident (51.2 MB each vs 192 MB L2).
// ---------------------------------------------------------------------------
__global__ void __launch_bounds__(256)
scatter_edges_kernel(const int* __restrict__ src, const int* __restrict__ dst,
                     const float* __restrict__ dinv, const float* __restrict__ h,
                     float* __restrict__ agg, int E) {
  int gid = blockIdx.x * blockDim.x + threadIdx.x;
  int e = gid >> 5;
  int lane = gid & 31;
  if (e >= E) return;
  int s = src[e];
  int d = dst[e];
  float norm = dinv[s] * dinv[d];
  float4 v = ((const float4*)(h + (size_t)s * HID))[lane];
  float* out = agg + (size_t)d * HID + lane * 4;
  atomicAdd(out + 0, v.x * norm);
  atomicAdd(out + 1, v.y * norm);
  atomicAdd(out + 2, v.z * norm);
  atomicAdd(out + 3, v.w * norm);
}

// ---------------------------------------------------------------------------
// Fused self-loop + bias + ReLU:  out = relu(agg + hlin*dinv^2 + b)
// (out may alias hlin: pure elementwise, same index read-then-write)
// ---------------------------------------------------------------------------
__global__ void finish_relu_kernel(const float* hlin, const float* agg,
                                   const float* __restrict__ dinv,
                                   const float* __restrict__ bias, float* out,
                                   int total) {
  int i = blockIdx.x * blockDim.x + threadIdx.x;
  if (i >= total) return;
  int node = i >> 7;  // /128
  int c = i & 127;
  float di = dinv[node];
  float v = agg[i] + hlin[i] * (di * di) + bias[c];
  out[i] = v > 0.0f ? v : 0.0f;
}

// ---------------------------------------------------------------------------
// Head: logits = tanh(X @ Wl + bl)  [N,8], scattered into per-graph pool sums.
// Wl is zero-padded 8->16 cols BY THE TDM (OOB columns read as zero).
// One wave per 16-row tile (WMMA f32).
// ---------------------------------------------------------------------------
__global__ void __launch_bounds__(256)
head_pool_kernel(const float* __restrict__ X, const float* __restrict__ Wl,
                 const float* __restrict__ bl, const int* __restrict__ batch,
                 float* __restrict__ pool, int N) {
  __shared__ float sB[128 * 16];  // 8 KB; cols 8..15 zero-filled by TDM OOB
  const int t = threadIdx.x;
  if (t == 0) {
    // tensor is 128 rows x 8 cols (stride 8); tile is 128 x 16 -> zero pad
    tdm_load_2d_f32((unsigned)(uintptr_t)sB, Wl, ACT, 128, 16, 128, ACT);
    __builtin_amdgcn_s_wait_tensorcnt(0);
  }
  __syncthreads();

  const int wave = t >> 5;
  const int lane = t & 31;
  const int tile = blockIdx.x * 8 + wave;
  const int rowbase = tile * 16;
  if (rowbase >= N) return;  // wave-uniform exit

  const int r = lane & 15;
  const int khalf = (lane >> 4) << 1;
  int row = rowbase + r;
  row = row < N ? row : N - 1;  // clamp; atomics below guard rr < N
  const float* xrow = X + (size_t)row * HID;

  v8f acc = {};
#pragma unroll 8
  for (int k = 0; k < 128; k += 4) {
    const int kk = k + khalf;
    float2 av = *(const float2*)(xrow + kk);
    v2f a;
    a.x = av.x;
    a.y = av.y;
    v2f b;
    b.x = sB[kk * 16 + r];
    b.y = sB[(kk + 1) * 16 + r];
    acc = wmma_f32_4(a, b, acc);
  }

  if (r < ACT) {
    const float bb = bl[r];
    const int mbase = rowbase + ((lane >> 4) << 3);
#pragma unroll
    for (int j = 0; j < 8; ++j) {
      int rr = mbase + j;
      if (rr < N) {
        float v = tanhf(acc[j] + bb);
        atomicAdd(&pool[batch[rr] * ACT + r], v);
      }
    }
  }
}

__global__ void divide_pool_kernel(const float* __restrict__ pool,
                                   const float* __restrict__ cnt,
                                   float* __restrict__ out, int total) {
  int i = blockIdx.x * blockDim.x + threadIdx.x;
  if (i < total) out[i] = pool[i] / fmaxf(cnt[i >> 3], 1.0f);
}

// ---------------------------------------------------------------------------
// Host launch
// ---------------------------------------------------------------------------
extern "C" void kernel_launch(void* const* d_in, const int* in_sizes, int n_in,
                              void* d_out, int out_size, void* d_ws,
                              size_t ws_size, hipStream_t stream) {
  const float* x = (const float*)d_in[0];
  const int* ei = (const int*)d_in[1];  // [2,E] row-major: src then dst
  const int* batch = (const int*)d_in[2];
  const float* W1 = (const float*)d_in[3];
  const float* b1 = (const float*)d_in[4];
  const float* W2 = (const float*)d_in[5];
  const float* b2 = (const float*)d_in[6];
  const float* W3 = (const float*)d_in[7];
  const float* b3 = (const float*)d_in[8];
  const float* Wl = (const float*)d_in[9];
  const float* bl = (const float*)d_in[10];
  float* out = (float*)d_out;

  const int N = in_sizes[0] / OBS;
  const int E = in_sizes[1] / 2;
  const int* src = ei;
  const int* dst = ei + E;

  // Workspace: two ping-pong node buffers + dinv + pool + counts
  float* bufA = (float*)d_ws;
  float* bufB = bufA + (size_t)N * HID;
  float* dinv = bufB + (size_t)N * HID;
  float* pool = dinv + N;
  float* cnt = pool + NGRAPHS * ACT;

  const int T = 256;
  const int total = N * HID;

  // --- symmetric normalization coefficients ---
  fill_one_kernel<<<(N + T - 1) / T, T, 0, stream>>>(dinv, N);
  count_deg_kernel<<<(E + T - 1) / T, T, 0, stream>>>(dst, dinv, E);
  deg_rsqrt_kernel<<<(N + T - 1) / T, T, 0, stream>>>(dinv, N);

  // --- three GCN layers (ping-pong bufA/bufB) ---
  const float* cur = x;
  float* lin = bufB;
  float* agg = bufA;
  const float* Ws[3] = {W1, W2, W3};
  const float* bs[3] = {b1, b2, b3};
  for (int l = 0; l < 3; ++l) {
    gemm128_wmma<<<(N + 127) / 128, T, 0, stream>>>(cur, Ws[l], lin, N);
    zero_kernel<<<(total + T - 1) / T, T, 0, stream>>>(agg, total);
    {
      long long nthreads = (long long)E * 32;
      scatter_edges_kernel<<<(int)((nthreads + T - 1) / T), T, 0, stream>>>(
          src, dst, dinv, lin, agg, E);
    }
    finish_relu_kernel<<<(total + T - 1) / T, T, 0, stream>>>(lin, agg, dinv,
                                                              bs[l], lin, total);
    cur = lin;
    float* tmp = lin;
    lin = agg;
    agg = tmp;
  }

  // --- head + global mean pool ---
  zero_kernel<<<(NGRAPHS * ACT + T - 1) / T, T, 0, stream>>>(pool,
                                                             NGRAPHS * ACT);
  zero_kernel<<<(NGRAPHS + T - 1) / T, T, 0, stream>>>(cnt, NGRAPHS);
  count_nodes_kernel<<<(N + T - 1) / T, T, 0, stream>>>(batch, cnt, N);
  {
    int tiles = (N + 15) / 16;
    head_pool_kernel<<<(tiles + 7) / 8, T, 0, stream>>>(cur, Wl, bl, batch,
                                                        pool, N);
  }
  divide_pool_kernel<<<(NGRAPHS * ACT + T - 1) / T, T, 0, stream>>>(
      pool, cnt, out, NGRAPHS * ACT);
}